// QloraMLPDebug_14121852469694
// MI455X (gfx1250) — compile-verified
//
#include <hip/hip_runtime.h>
#include <hip/hip_bf16.h>

typedef float v2f __attribute__((ext_vector_type(2)));
typedef float v8f __attribute__((ext_vector_type(8)));

#define D_MODEL   4096
#define D4        1024      // D_MODEL/4  (w1q/w3q row length)
#define HID       11008
#define H4        2752      // HID/4      (w2q row length)
#define M_ROWS    16
#define LORA_SCALE 0.5f

// ---------------- ws layout (float offsets) ----------------
#define XF_OFF    0                         // 16 x 1024
#define T1_OFF    (XF_OFF + M_ROWS*D4)      // 16 x 2
#define T3_OFF    (T1_OFF + 32)             // 16 x 2
#define T2_OFF    (T3_OFF + 32)             // 16 x 2
#define HOUT_OFF  (T2_OFF + 32)             // 16 x 11008
#define HF_OFF    (HOUT_OFF + M_ROWS*HID)   // 16 x 2752
// total = 236640 floats = 946560 bytes

// ---------------------------------------------------------------------------
// fold x: xf[m][j] = sum_{c<4} x[m][c*1024 + j]
__global__ void qlora_fold_x(const float* __restrict__ x, float* __restrict__ xf) {
    int idx = blockIdx.x * blockDim.x + threadIdx.x;     // 0 .. 16*1024-1
    if (idx >= M_ROWS * D4) return;
    int m = idx / D4, j = idx % D4;
    const float* xr = x + (size_t)m * D_MODEL;
    xf[idx] = xr[j] + xr[D4 + j] + xr[2*D4 + j] + xr[3*D4 + j];
}

// fold h: hf[m][j] = sum_{c<4} hout[m][c*2752 + j]
__global__ void qlora_fold_h(const float* __restrict__ hout, float* __restrict__ hf) {
    int idx = blockIdx.x * blockDim.x + threadIdx.x;     // 0 .. 16*2752-1
    if (idx >= M_ROWS * H4) return;
    int m = idx / H4, j = idx % H4;
    const float* hr = hout + (size_t)m * HID;
    hf[idx] = hr[j] + hr[H4 + j] + hr[2*H4 + j] + hr[3*H4 + j];
}

// ---------------------------------------------------------------------------
// t[m][r] = dot(X[m, 0:n], A[r, 0:n]); one block per (m,r) pair, LDS tree reduce
__global__ __launch_bounds__(256) void qlora_rowdot(const float* __restrict__ X,
                                                    const float* __restrict__ A,
                                                    float* __restrict__ t, int n) {
    __shared__ float red[256];
    int m = blockIdx.x >> 1;
    int r = blockIdx.x & 1;
    const float* xr = X + (size_t)m * n;
    const float* ar = A + (size_t)r * n;
    float acc = 0.f;
    for (int i = threadIdx.x; i < n; i += 256)
        acc += xr[i] * ar[i];
    red[threadIdx.x] = acc;
    __syncthreads();
    for (int s = 128; s > 0; s >>= 1) {
        if (threadIdx.x < s) red[threadIdx.x] += red[threadIdx.x + s];
        __syncthreads();
    }
    if (threadIdx.x == 0) t[m * 2 + r] = red[0];
}

// ---------------------------------------------------------------------------
// gate/up: one wave per 16-wide output tile of the hidden dim.
// c1 = xf @ w1q^T tile, c3 = xf @ w3q^T tile via V_WMMA_F32_16X16X4_F32,
// then + rank-2 LoRA, h = silu(g)*u  -> hout (16 x 11008)
//
// WMMA f32 16x16x4 striping (wave32):
//   A (16x4):  lanes 0-15 hold M=q, K={0,1}; lanes 16-31 hold M=q, K={2,3}
//   B (4x16):  lanes 0-15 hold N=q, K={0,1}; lanes 16-31 hold N=q, K={2,3}
// -> both A and B per lane are a contiguous float2 at (row, k0 + 2*half):
//    no shuffles needed, each half-wave loads complementary 8B of the 16B chunk.
__global__ __launch_bounds__(32) void qlora_gateup(const float* __restrict__ xf,
                                                   const float* __restrict__ w1q,
                                                   const float* __restrict__ w3q,
                                                   const float* __restrict__ b1,
                                                   const float* __restrict__ b3,
                                                   const float* __restrict__ t1,
                                                   const float* __restrict__ t3,
                                                   float* __restrict__ hout) {
    const int lane = threadIdx.x & 31;
    const int half = lane >> 4;          // 0: K=0..1 , 1: K=2..3 (A & B striping)
    const int q    = lane & 15;          // M row (for A) / N col (for B,C)
    const int o    = blockIdx.x * 16 + q;

    const float* w1row = w1q + (size_t)o * D4 + 2 * half;
    const float* w3row = w3q + (size_t)o * D4 + 2 * half;
    const float* xrow  = xf  + (size_t)q * D4 + 2 * half;

    v8f c1 = {};
    v8f c3 = {};
    #pragma unroll 8
    for (int k0 = 0; k0 < D4; k0 += 4) {
        v2f a   = *(const v2f*)(xrow  + k0);   // A: K pair for this half
        v2f bb1 = *(const v2f*)(w1row + k0);   // B: exact 8B this lane owns
        v2f bb3 = *(const v2f*)(w3row + k0);
        c1 = __builtin_amdgcn_wmma_f32_16x16x4_f32(false, a, false, bb1,
                                                   (short)0, c1, false, false);
        c3 = __builtin_amdgcn_wmma_f32_16x16x4_f32(false, a, false, bb3,
                                                   (short)0, c3, false, false);
    }

    const float b1v0 = b1[o * 2 + 0], b1v1 = b1[o * 2 + 1];
    const float b3v0 = b3[o * 2 + 0], b3v1 = b3[o * 2 + 1];
    #pragma unroll
    for (int i = 0; i < 8; ++i) {
        int m = i + 8 * half;            // C layout: VGPR i holds M=i (lo) / M=8+i (hi)
        float g = c1[i] + LORA_SCALE * (t1[m * 2 + 0] * b1v0 + t1[m * 2 + 1] * b1v1);
        float u = c3[i] + LORA_SCALE * (t3[m * 2 + 0] * b3v0 + t3[m * 2 + 1] * b3v1);
        float s = g / (1.0f + __expf(-g));   // silu
        hout[(size_t)m * HID + o] = s * u;
    }
}

// ---------------------------------------------------------------------------
// out = hf @ w2q^T + lora2 : one wave per 16-wide tile of d_model
__global__ __launch_bounds__(32) void qlora_down(const float* __restrict__ hf,
                                                 const float* __restrict__ w2q,
                                                 const float* __restrict__ b2,
                                                 const float* __restrict__ t2,
                                                 float* __restrict__ out) {
    const int lane = threadIdx.x & 31;
    const int half = lane >> 4;
    const int q    = lane & 15;
    const int j    = blockIdx.x * 16 + q;

    const float* wrow = w2q + (size_t)j * H4 + 2 * half;
    const float* hrow = hf  + (size_t)q * H4 + 2 * half;

    v8f c = {};
    #pragma unroll 8
    for (int k0 = 0; k0 < H4; k0 += 4) {
        v2f a  = *(const v2f*)(hrow + k0);
        v2f bb = *(const v2f*)(wrow + k0);
        c = __builtin_amdgcn_wmma_f32_16x16x4_f32(false, a, false, bb,
                                                  (short)0, c, false, false);
    }

    const float b2v0 = b2[j * 2 + 0], b2v1 = b2[j * 2 + 1];
    #pragma unroll
    for (int i = 0; i < 8; ++i) {
        int m = i + 8 * half;
        out[(size_t)m * D_MODEL + j] =
            c[i] + LORA_SCALE * (t2[m * 2 + 0] * b2v0 + t2[m * 2 + 1] * b2v1);
    }
}

// ---------------------------------------------------------------------------
extern "C" void kernel_launch(void* const* d_in, const int* in_sizes, int n_in,
                              void* d_out, int out_size, void* d_ws, size_t ws_size,
                              hipStream_t stream) {
    const float* x   = (const float*)d_in[0];
    const float* w1q = (const float*)d_in[1];
    const float* a1  = (const float*)d_in[2];
    const float* b1  = (const float*)d_in[3];
    const float* w3q = (const float*)d_in[4];
    const float* a3  = (const float*)d_in[5];
    const float* b3  = (const float*)d_in[6];
    const float* w2q = (const float*)d_in[7];
    const float* a2  = (const float*)d_in[8];
    const float* b2  = (const float*)d_in[9];
    float* out = (float*)d_out;

    float* ws   = (float*)d_ws;
    float* xf   = ws + XF_OFF;
    float* t1   = ws + T1_OFF;
    float* t3   = ws + T3_OFF;
    float* t2   = ws + T2_OFF;
    float* hout = ws + HOUT_OFF;
    float* hf   = ws + HF_OFF;

    // 1) fold x
    qlora_fold_x<<<(M_ROWS * D4 + 255) / 256, 256, 0, stream>>>(x, xf);
    // 2) LoRA left projections t1, t3 (16x2 each)
    qlora_rowdot<<<32, 256, 0, stream>>>(x, a1, t1, D_MODEL);
    qlora_rowdot<<<32, 256, 0, stream>>>(x, a3, t3, D_MODEL);
    // 3) gate/up WMMA GEMMs + silu*u  (688 tiles of 16)
    qlora_gateup<<<HID / 16, 32, 0, stream>>>(xf, w1q, w3q, b1, b3, t1, t3, hout);
    // 4) fold h
    qlora_fold_h<<<(M_ROWS * H4 + 255) / 256, 256, 0, stream>>>(hout, hf);
    // 5) LoRA left projection t2
    qlora_rowdot<<<32, 256, 0, stream>>>(hout, a2, t2, HID);
    // 6) down-projection WMMA GEMM + LoRA  (256 tiles of 16)
    qlora_down<<<D_MODEL / 16, 32, 0, stream>>>(hf, w2q, b2, t2, out);
}